// GraphConvolution_25950192402741
// MI455X (gfx1250) — compile-verified
//
#include <hip/hip_runtime.h>
#include <hip/hip_bf16.h>

typedef float v2f __attribute__((ext_vector_type(2)));
typedef float v8f __attribute__((ext_vector_type(8)));

#define N_NODES 65536
#define DIM 64
#define BPITCH 68   // LDS pitch for transposed ortho: bank = (4n+k)%64, conflict-free

// ---------------------------------------------------------------------------
// Kernel 1: Cayley transform  ortho = solve(I+S, (I-S)^T)^T,  S = W - W^T
// One workgroup, Gauss-Jordan in LDS (I+S is well conditioned; no pivoting).
// ---------------------------------------------------------------------------
__global__ void cayley_kernel(const float* __restrict__ W, float* __restrict__ ortho) {
    __shared__ float A[DIM * DIM];   // I + S
    __shared__ float B[DIM * DIM];   // (I - S)^T, becomes X = A^{-1} B
    __shared__ float fac[DIM];

    const int tid = threadIdx.x;

    // Build A and B faithfully from the reference formulas.
    for (int idx = tid; idx < DIM * DIM; idx += blockDim.x) {
        const int i = idx >> 6, j = idx & 63;
        const float wij = W[i * DIM + j];
        const float wji = W[j * DIM + i];
        const float sij = wij - wji;             // S[i][j]
        const float sji = wji - wij;             // S[j][i]
        const float eye = (i == j) ? 1.0f : 0.0f;
        A[idx] = eye + sij;                      // (I + S)[i][j]
        B[idx] = eye - sji;                      // (I - S)^T [i][j] = I[i][j] - S[j][i]
    }
    __syncthreads();

    // Gauss-Jordan elimination: A -> I, B -> A^{-1} B
    for (int k = 0; k < DIM; ++k) {
        const float invp = 1.0f / A[k * DIM + k];
        __syncthreads();
        if (tid < DIM) {
            A[k * DIM + tid] *= invp;
            B[k * DIM + tid] *= invp;
        }
        __syncthreads();
        if (tid < DIM) fac[tid] = (tid == k) ? 0.0f : A[tid * DIM + k];
        __syncthreads();
        for (int idx = tid; idx < DIM * DIM; idx += blockDim.x) {
            const int i = idx >> 6, j = idx & 63;
            const float f = fac[i];
            A[idx] -= f * A[k * DIM + j];
            B[idx] -= f * B[k * DIM + j];
        }
        __syncthreads();
    }

    // ortho = X^T
    for (int idx = tid; idx < DIM * DIM; idx += blockDim.x) {
        const int i = idx >> 6, j = idx & 63;
        ortho[idx] = B[j * DIM + i];
    }
}

// ---------------------------------------------------------------------------
// Kernel 2: support = x @ ortho using V_WMMA_F32_16X16X4_F32 (full fp32).
// Block = 256 threads = 8 waves; each wave computes a 16x64 strip.
// A (16x4 f32): lanes 0-15 hold M=lane, K={k0,k0+1}; lanes 16-31 K={k0+2,k0+3}.
// B (4x16 f32): VGPR0 = row k0 (lanes 0-15) / row k0+2 (lanes 16-31); VGPR1 = k0+1 / k0+3.
// C (16x16 f32): VGPR v -> M=v (lanes 0-15), M=v+8 (lanes 16-31), N=lane&15.
// ortho is staged TRANSPOSED in LDS (pitch 68) so each B fragment {K,K+1} for a
// fixed column is one naturally-paired ds_load_b64 -> no VGPR repacking movs.
// ---------------------------------------------------------------------------
__global__ void gemm_wmma_kernel(const float* __restrict__ x,
                                 const float* __restrict__ ortho,
                                 float* __restrict__ support) {
    __shared__ float bsh[DIM * BPITCH];   // ortho^T, padded (17 KB)

    const int tid  = threadIdx.x;
    const int wave = tid >> 5;
    const int lane = tid & 31;
    const int lh   = lane >> 4;        // 0 or 1 (lane half)
    const int ll   = lane & 15;

    // Stage ortho transposed: bsh[n*BPITCH + k] = ortho[k][n]
    for (int i = tid; i < DIM * DIM; i += 256) {
        const int k = i >> 6, n = i & 63;
        bsh[n * BPITCH + k] = ortho[i];
    }
    __syncthreads();

    const int rowBase = (blockIdx.x * 8 + wave) * 16;
    const float* __restrict__ xrow = x + (size_t)(rowBase + ll) * DIM;

    // Per-lane base for B fragments: column n = nt*16 + ll, k offset lh*2
    const float* __restrict__ b0p = &bsh[(0 * 16 + ll) * BPITCH + lh * 2];
    const float* __restrict__ b1p = &bsh[(1 * 16 + ll) * BPITCH + lh * 2];
    const float* __restrict__ b2p = &bsh[(2 * 16 + ll) * BPITCH + lh * 2];
    const float* __restrict__ b3p = &bsh[(3 * 16 + ll) * BPITCH + lh * 2];

    v8f acc0 = {}, acc1 = {}, acc2 = {}, acc3 = {};

#pragma unroll
    for (int k0 = 0; k0 < DIM; k0 += 4) {
        // A fragment: 2 consecutive fp32 per lane (8B aligned global load)
        const v2f a  = *(const v2f*)(xrow + k0 + lh * 2);
        // B fragments: {ortho[k][n], ortho[k+1][n]} adjacent in transposed LDS
        const v2f b0 = *(const v2f*)(b0p + k0);
        const v2f b1 = *(const v2f*)(b1p + k0);
        const v2f b2 = *(const v2f*)(b2p + k0);
        const v2f b3 = *(const v2f*)(b3p + k0);

        acc0 = __builtin_amdgcn_wmma_f32_16x16x4_f32(false, a, false, b0, (short)0, acc0, false, false);
        acc1 = __builtin_amdgcn_wmma_f32_16x16x4_f32(false, a, false, b1, (short)0, acc1, false, false);
        acc2 = __builtin_amdgcn_wmma_f32_16x16x4_f32(false, a, false, b2, (short)0, acc2, false, false);
        acc3 = __builtin_amdgcn_wmma_f32_16x16x4_f32(false, a, false, b3, (short)0, acc3, false, false);
    }

    // Store: C VGPR v holds row (rowBase + v + lh*8), col = ntile*16 + ll
#pragma unroll
    for (int v = 0; v < 8; ++v) {
        float* orow = support + (size_t)(rowBase + v + lh * 8) * DIM + ll;
        orow[0 * 16] = acc0[v];
        orow[1 * 16] = acc1[v];
        orow[2 * 16] = acc2[v];
        orow[3 * 16] = acc3[v];
    }
}

// ---------------------------------------------------------------------------
// Kernel 3: out = x_0 + bias (GAMMA*agg accumulated on top by kernel 4)
// ---------------------------------------------------------------------------
__global__ void init_out_kernel(const float* __restrict__ x0,
                                const float* __restrict__ bias,
                                float* __restrict__ out, int total4) {
    const int idx = blockIdx.x * blockDim.x + threadIdx.x;
    if (idx >= total4) return;
    const float4 v = ((const float4*)x0)[idx];
    const float4 b = ((const float4*)bias)[idx & 15];   // (element_base & 63) / 4
    float4 r;
    r.x = v.x + b.x; r.y = v.y + b.y; r.z = v.z + b.z; r.w = v.w + b.w;
    ((float4*)out)[idx] = r;
}

// ---------------------------------------------------------------------------
// Kernel 4: COO scatter-add.  One wave per edge; lane owns features 2l,2l+1.
// out[row] += edge_val * support[col]   (GAMMA == 1)
// ---------------------------------------------------------------------------
__global__ void edge_scatter_kernel(const float* __restrict__ support,
                                    const float* __restrict__ edge_val,
                                    const int* __restrict__ edge_row,
                                    const int* __restrict__ edge_col,
                                    float* __restrict__ out, int E) {
    const int wave   = (blockIdx.x * blockDim.x + threadIdx.x) >> 5;
    const int lane   = threadIdx.x & 31;
    const int nwaves = (gridDim.x * blockDim.x) >> 5;

    for (int e = wave; e < E; e += nwaves) {
        const int   row = edge_row[e];       // uniform across the wave -> scalar
        const int   col = edge_col[e];
        const float val = edge_val[e];
        const float2 f  = *(const float2*)(support + (size_t)col * DIM + lane * 2);
        float* dst = out + (size_t)row * DIM + lane * 2;
        unsafeAtomicAdd(dst + 0, val * f.x);   // global_atomic_add_f32
        unsafeAtomicAdd(dst + 1, val * f.y);
    }
}

// ---------------------------------------------------------------------------
// Launch: inputs are {x, x_0, edge_val, weight, bias, edge_row, edge_col}
// ---------------------------------------------------------------------------
extern "C" void kernel_launch(void* const* d_in, const int* in_sizes, int n_in,
                              void* d_out, int out_size, void* d_ws, size_t ws_size,
                              hipStream_t stream) {
    const float* x        = (const float*)d_in[0];
    const float* x0       = (const float*)d_in[1];
    const float* edge_val = (const float*)d_in[2];
    const float* weight   = (const float*)d_in[3];
    const float* bias     = (const float*)d_in[4];
    const int*   edge_row = (const int*)d_in[5];
    const int*   edge_col = (const int*)d_in[6];
    float*       out      = (float*)d_out;

    const int n = in_sizes[0] / DIM;     // 65536 nodes
    const int E = in_sizes[2];           // 1048576 edges

    // Workspace: [ortho 64*64 f32 | support n*64 f32]
    float* ortho   = (float*)d_ws;
    float* support = ortho + DIM * DIM;

    // 1) Cayley transform (single workgroup)
    cayley_kernel<<<1, 256, 0, stream>>>(weight, ortho);

    // 2) support = x @ ortho via fp32 WMMA (8 waves/block, 128 rows/block)
    gemm_wmma_kernel<<<n / 128, 256, 0, stream>>>(x, ortho, support);

    // 3) out = x_0 + bias
    const int total4 = n * DIM / 4;
    init_out_kernel<<<(total4 + 255) / 256, 256, 0, stream>>>(x0, bias, out, total4);

    // 4) out += edge_val * support[edge_col] scattered to edge_row
    edge_scatter_kernel<<<1024, 256, 0, stream>>>(support, edge_val, edge_row,
                                                  edge_col, out, E);
}